// GeoFeaturizer_28046136443287
// MI455X (gfx1250) — compile-verified
//
#include <hip/hip_runtime.h>
#include <math.h>

#define NUM_BATCH 8
#define NUM_GLOBAL 3
#define EPSF 1e-20f

typedef unsigned int u32x4 __attribute__((ext_vector_type(4)));
typedef int i32x4 __attribute__((ext_vector_type(4)));
typedef int i32x8 __attribute__((ext_vector_type(8)));

// decouple(): direct = u/(|u|+1e-6), then 16 RBF features of |u|.
// mu = linspace(0,20,16) -> step 20/15 ; sigma = 20/16 = 1.25
__device__ __forceinline__ void decouple3(float ux, float uy, float uz, float* __restrict__ o) {
  float sq = ux * ux + uy * uy + uz * uz;
  float norm = sqrtf(sq);                 // sq==0 -> 0, matches reference where()
  float inv = 1.0f / (norm + 1e-6f);
  o[0] = ux * inv; o[1] = uy * inv; o[2] = uz * inv;
  const float sigma_inv = 1.0f / 1.25f;
  #pragma unroll
  for (int k = 0; k < 16; ++k) {
    float mu = (20.0f / 15.0f) * (float)k;
    float z = (norm - mu) * sigma_inv;
    o[3 + k] = expf(-z * z);
  }
}

// ---------------- Kernel 1: per-node frames + V features ----------------
__global__ void k_nodes(const float* __restrict__ X,
                        const int* __restrict__ batch_id,
                        const int* __restrict__ chain_enc,
                        float* __restrict__ Vout,   // [NV,76]
                        float* __restrict__ Rout,   // [NV,9]
                        float* __restrict__ tout,   // [NV,3]
                        float* __restrict__ bout,   // [NV]
                        float* __restrict__ cout,   // [NV]
                        int N) {
  int n = blockIdx.x * blockDim.x + threadIdx.x;
  if (n >= N) return;
  const float* x = X + (size_t)n * 12;
  float cax = x[3], cay = x[4], caz = x[5];
  float nlx = x[0] - cax, nly = x[1] - cay, nlz = x[2] - caz;
  float cx = x[6] - cax, cy = x[7] - cay, cz = x[8] - caz;

  float n1 = sqrtf(EPSF + cx * cx + cy * cy);
  float s1 = -cy / n1, c1 = cx / n1;
  float n2 = sqrtf(EPSF + cx * cx + cy * cy + cz * cz);
  float s2 = cz / n2, c2 = sqrtf(cx * cx + cy * cy) / n2;
  // Rc = R2*R1
  float Rc00 =  c2 * c1, Rc01 = -c2 * s1, Rc02 = s2;
  float Rc10 =  s1,      Rc11 =  c1,      Rc12 = 0.0f;
  float Rc20 = -s2 * c1, Rc21 =  s2 * s1, Rc22 = c2;
  // nr = Rc * nl  (need components 1,2)
  float nr1 = Rc10 * nlx + Rc11 * nly + Rc12 * nlz;
  float nr2 = Rc20 * nlx + Rc21 * nly + Rc22 * nlz;
  float n3 = sqrtf(EPSF + nr1 * nr1 + nr2 * nr2);
  float sn = -nr2 / n3, cn = nr1 / n3;
  // M = Rn * Rc ; stored R = M^T
  float M00 = Rc00, M01 = Rc01, M02 = Rc02;
  float M10 = cn * Rc10 - sn * Rc20, M11 = cn * Rc11 - sn * Rc21, M12 = cn * Rc12 - sn * Rc22;
  float M20 = sn * Rc10 + cn * Rc20, M21 = sn * Rc11 + cn * Rc21, M22 = sn * Rc12 + cn * Rc22;
  float R[9] = { M00, M10, M20,  M01, M11, M21,  M02, M12, M22 };

  float* rr = Rout + (size_t)n * 9;
  #pragma unroll
  for (int i = 0; i < 9; ++i) rr[i] = R[i];
  float* tr = tout + (size_t)n * 3;
  tr[0] = cax; tr[1] = cay; tr[2] = caz;
  bout[n] = (float)batch_id[n];
  cout[n] = (float)chain_enc[n];

  // diffX rows: Xflat[n*4+a] - Xflat[n*4+a-1], flat row 0 is zero
  float row[76];
  #pragma unroll
  for (int a = 0; a < 4; ++a) {
    float ux = 0.0f, uy = 0.0f, uz = 0.0f;
    size_t fi = (size_t)n * 4 + (size_t)a;
    if (fi > 0) {
      const float* cur = X + fi * 3;
      ux = cur[0] - cur[-3];
      uy = cur[1] - cur[-2];
      uz = cur[2] - cur[-1];
    }
    // p = R^T u
    float px = R[0] * ux + R[3] * uy + R[6] * uz;
    float py = R[1] * ux + R[4] * uy + R[7] * uz;
    float pz = R[2] * ux + R[5] * uy + R[8] * uz;
    decouple3(px, py, pz, row + a * 19);
  }
  #pragma unroll
  for (int i = 0; i < 76; ++i) { if (row[i] != row[i]) row[i] = 0.0f; }
  float4* vd = (float4*)(Vout + (size_t)n * 76);
  const float4* vs = (const float4*)row;
  #pragma unroll
  for (int i = 0; i < 19; ++i) vd[i] = vs[i];
}

// ---------------- Kernel 2: global-node tail rows ----------------
__global__ void k_global_nodes(float* __restrict__ Vout, float* __restrict__ Rout,
                               float* __restrict__ tout, float* __restrict__ bout,
                               float* __restrict__ cout, int N) {
  int g = threadIdx.x;
  if (g >= NUM_GLOBAL * NUM_BATCH) return;
  int node = N + g;
  float d = (float)(g % NUM_GLOBAL);
  float row[76];
  #pragma unroll
  for (int j = 0; j < 38; ++j) {
    float freq = expf((float)(2 * j) * (-9.210340371976184f / 76.0f));
    float ang = d * freq;
    row[j]      = cosf(ang);
    row[38 + j] = sinf(ang);
  }
  float* vr = Vout + (size_t)node * 76;
  #pragma unroll
  for (int i = 0; i < 76; ++i) vr[i] = row[i];
  float* rr = Rout + (size_t)node * 9;
  rr[0] = 1.0f; rr[1] = 0.0f; rr[2] = 0.0f;
  rr[3] = 0.0f; rr[4] = 1.0f; rr[5] = 0.0f;
  rr[6] = 0.0f; rr[7] = 0.0f; rr[8] = 1.0f;
  float* tr = tout + (size_t)node * 3;
  tr[0] = 0.0f; tr[1] = 0.0f; tr[2] = 0.0f;
  bout[node] = (float)(g % NUM_BATCH);
  cout[node] = 1001.0f;
}

// ---------------- Kernel 3: per-edge features ----------------
__global__ void k_edges(const float* __restrict__ X, const int* __restrict__ eidx,
                        const float* __restrict__ Rall, const float* __restrict__ tall,
                        float* __restrict__ Eout,    // [NEa,196]
                        float* __restrict__ RtsOut,  // [NEa,9]
                        float* __restrict__ ttsOut,  // [NEa,3]
                        float* __restrict__ edgeOut, // [2,NEa]
                        int E, long long NEa) {
  int e = blockIdx.x * blockDim.x + threadIdx.x;
  if (e >= E) return;
  int s = eidx[e];
  int d = eidx[(size_t)E + e];
  __builtin_prefetch(Rall + (size_t)s * 9, 0, 0);
  __builtin_prefetch(Rall + (size_t)d * 9, 0, 0);
  __builtin_prefetch(X + (size_t)s * 12, 0, 0);
  __builtin_prefetch(X + (size_t)d * 12, 0, 0);

  float Rs[9], Rd[9], ts3[3], td3[3];
  #pragma unroll
  for (int i = 0; i < 9; ++i) { Rs[i] = Rall[(size_t)s * 9 + i]; Rd[i] = Rall[(size_t)d * 9 + i]; }
  #pragma unroll
  for (int i = 0; i < 3; ++i) { ts3[i] = tall[(size_t)s * 3 + i]; td3[i] = tall[(size_t)d * 3 + i]; }

  float Rts[9];
  #pragma unroll
  for (int i = 0; i < 3; ++i)
    #pragma unroll
    for (int k = 0; k < 3; ++k)
      Rts[i * 3 + k] = Rd[0 + i] * Rs[0 + k] + Rd[3 + i] * Rs[3 + k] + Rd[6 + i] * Rs[6 + k];
  float dt0 = ts3[0] - td3[0], dt1 = ts3[1] - td3[1], dt2 = ts3[2] - td3[2];
  float tts[3];
  #pragma unroll
  for (int i = 0; i < 3; ++i)
    tts[i] = Rd[0 + i] * dt0 + Rd[3 + i] * dt1 + Rd[6 + i] * dt2;

  float* erow = Eout + (size_t)e * 196;
  #pragma unroll
  for (int a = 0; a < 8; ++a) {
    const float* ap = (a < 4) ? (X + (size_t)s * 12 + a * 3)
                              : (X + (size_t)d * 12 + (a - 4) * 3);
    float ux = ap[0] - ts3[0], uy = ap[1] - ts3[1], uz = ap[2] - ts3[2];
    float px = Rs[0] * ux + Rs[3] * uy + Rs[6] * uz;
    float py = Rs[1] * ux + Rs[4] * uy + Rs[7] * uz;
    float pz = Rs[2] * ux + Rs[5] * uy + Rs[8] * uz;
    float o[19];
    decouple3(px, py, pz, o);
    #pragma unroll
    for (int i = 0; i < 19; ++i) erow[a * 19 + i] = o[i];
  }
  #pragma unroll
  for (int i = 0; i < 3; ++i)
    #pragma unroll
    for (int k = 0; k < 3; ++k)
      erow[152 + i * 3 + k] = Rts[k * 3 + i];
  {
    float o[19];
    decouple3(tts[0], tts[1], tts[2], o);
    #pragma unroll
    for (int i = 0; i < 19; ++i) erow[161 + i] = o[i];
  }
  {
    float dp = (float)(s - d);
    #pragma unroll
    for (int j = 0; j < 8; ++j) {
      float freq = expf((float)(2 * j) * (-9.210340371976184f / 16.0f));
      float ang = dp * freq;
      erow[180 + j] = cosf(ang);
      erow[188 + j] = sinf(ang);
    }
  }
  float* rr = RtsOut + (size_t)e * 9;
  #pragma unroll
  for (int i = 0; i < 9; ++i) rr[i] = Rts[i];
  float* tr = ttsOut + (size_t)e * 3;
  tr[0] = tts[0]; tr[1] = tts[1]; tr[2] = tts[2];
  edgeOut[e] = (float)s;
  edgeOut[NEa + e] = (float)d;
}

// ---------------- Kernel 4: TDM zero fill of the E-tail bulk ----------------
// One TENSOR_STORE_FROM_LDS per 64KB tile: D# = 1D tile, 16384 x 4B elements,
// sourced from a zeroed 64KB LDS buffer. Tracked by TENSORcnt.
#define TDM_TILE_ELTS 16384  // 64 KB of 4-byte elements (tile_dim0, fits u16)

__global__ void k_zero_tdm(float* __restrict__ base, int numTiles) {
  __shared__ float4 zbuf[TDM_TILE_ELTS / 4];  // 64 KB
  #pragma unroll
  for (int i = 0; i < TDM_TILE_ELTS / 4 / 256; ++i)
    zbuf[threadIdx.x + i * 256] = make_float4(0.0f, 0.0f, 0.0f, 0.0f);
  __syncthreads();
  if (threadIdx.x == 0) {
    unsigned lds_addr = (unsigned)(size_t)(&zbuf[0]);  // low 32 bits = LDS byte address
    for (int t = blockIdx.x; t < numTiles; t += gridDim.x) {
      unsigned long long gaddr =
          (unsigned long long)(size_t)base + (unsigned long long)t * (TDM_TILE_ELTS * 4ull);
      // D# group 0: count=1 | lds_addr | global_addr[56:0] | type=2
      u32x4 g0;
      g0[0] = 1u;
      g0[1] = lds_addr;
      g0[2] = (unsigned)gaddr;
      g0[3] = ((unsigned)(gaddr >> 32) & 0x01FFFFFFu) | (2u << 30);
      // D# group 1: data_size=4B (code 2); tensor_dim0 = tile_dim0 = TDM_TILE_ELTS;
      // tensor_dim1 = 1; tile_dim1 = tile_dim2 = 0 (1D tile); benign strides.
      i32x8 g1;
      g1[0] = (int)(2u << 16);                               // workgroup_mask=0, data_size=2
      g1[1] = (int)(((unsigned)TDM_TILE_ELTS & 0xFFFFu) << 16);        // tensor_dim0 lo16
      g1[2] = (int)((((unsigned)TDM_TILE_ELTS >> 16) & 0xFFFFu) | (1u << 16)); // dim0 hi16 | tensor_dim1 lo16
      g1[3] = (int)(((unsigned)TDM_TILE_ELTS & 0xFFFFu) << 16);        // dim1 hi16=0 | tile_dim0
      g1[4] = 0;                                             // tile_dim1=0, tile_dim2=0
      g1[5] = (int)(unsigned)TDM_TILE_ELTS;                  // tensor_dim0_stride lo32
      g1[6] = (int)(((unsigned)TDM_TILE_ELTS & 0xFFFFu) << 16); // stride0 hi16=0 | stride1 lo16
      g1[7] = 0;                                             // stride1 hi32
      i32x4 gz = { 0, 0, 0, 0 };
#if defined(__clang_major__) && (__clang_major__ >= 23)
      i32x8 gz8 = { 0, 0, 0, 0, 0, 0, 0, 0 };
      __builtin_amdgcn_tensor_store_from_lds(g0, g1, gz, gz, gz8, 0);
#else
      __builtin_amdgcn_tensor_store_from_lds(g0, g1, gz, gz, 0);
#endif
    }
    __builtin_amdgcn_s_wait_tensorcnt(0);
  }
}

// ---------------- Kernel 5: async LDS->global zero fill (remainder) ----------------
__global__ void k_zero_async(float* __restrict__ base, long long total_f4) {
  __shared__ float4 zbuf[256];
  zbuf[threadIdx.x] = make_float4(0.0f, 0.0f, 0.0f, 0.0f);
  __syncthreads();
  unsigned lds_off = (unsigned)(size_t)(&zbuf[threadIdx.x]);
  long long tid = (long long)blockIdx.x * blockDim.x + threadIdx.x;
  long long stride = (long long)gridDim.x * blockDim.x;
  for (long long i = tid; i < total_f4; i += stride) {
    unsigned long long gaddr = (unsigned long long)(size_t)(base) + (unsigned long long)i * 16ull;
    asm volatile("global_store_async_from_lds_b128 %0, %1, off"
                 :: "v"(gaddr), "v"(lds_off) : "memory");
  }
  asm volatile("s_wait_asynccnt 0x0" ::: "memory");
}

// ---------------- Kernel 6: global-edge tail (Rts/tts/edge indices) ----------------
__global__ void k_global_edges(const int* __restrict__ batch_id,
                               float* __restrict__ RtsOut,
                               float* __restrict__ ttsOut, float* __restrict__ edgeOut,
                               int N, int E, long long NEa) {
  long long ge = (long long)blockIdx.x * blockDim.x + threadIdx.x;
  long long NEg = 2LL * NUM_GLOBAL * N;
  if (ge >= NEg) return;
  long long ri = (long long)E + ge;
  float* rr = RtsOut + (size_t)ri * 9;
  rr[0] = 1.0f; rr[1] = 0.0f; rr[2] = 0.0f;
  rr[3] = 0.0f; rr[4] = 1.0f; rr[5] = 0.0f;
  rr[6] = 0.0f; rr[7] = 0.0f; rr[8] = 1.0f;
  float* tr = ttsOut + (size_t)ri * 3;
  tr[0] = 0.0f; tr[1] = 0.0f; tr[2] = 0.0f;
  long long GN = (long long)NUM_GLOBAL * N;
  int srcv, dstv;
  if (ge < GN) {
    int k = (int)(ge / N); int i = (int)(ge % N);
    srcv = batch_id[i] + N + k * NUM_BATCH;
    dstv = i;
  } else {
    long long j = ge - GN;
    int k = (int)(j / N); int i = (int)(j % N);
    srcv = i;
    dstv = batch_id[i] + N + k * NUM_BATCH;
  }
  edgeOut[ri] = (float)srcv;
  edgeOut[NEa + ri] = (float)dstv;
}

extern "C" void kernel_launch(void* const* d_in, const int* in_sizes, int n_in,
                              void* d_out, int out_size, void* d_ws, size_t ws_size,
                              hipStream_t stream) {
  (void)n_in; (void)out_size; (void)d_ws; (void)ws_size;
  const float* X        = (const float*)d_in[0];
  const int*   eidx     = (const int*)d_in[1];
  const int*   batch_id = (const int*)d_in[2];
  const int*   chain    = (const int*)d_in[3];
  int N = in_sizes[2];          // batch_id length
  int E = in_sizes[1] / 2;      // edge_idx is [2,E]

  long long NV  = (long long)N + NUM_GLOBAL * NUM_BATCH;
  long long NEg = 2LL * NUM_GLOBAL * N;
  long long NEa = (long long)E + NEg;

  float* out  = (float*)d_out;
  float* V    = out;                    // [NV,76]
  float* Ef   = V    + NV * 76;         // [NEa,196]
  float* Rall = Ef   + NEa * 196;       // [NV,9]
  float* tall = Rall + NV * 9;          // [NV,3]
  float* Rts  = tall + NV * 3;          // [NEa,9]
  float* tts  = Rts  + NEa * 9;         // [NEa,3]
  float* ball = tts  + NEa * 3;         // [NV]
  float* edge = ball + NV;              // [2,NEa]
  float* call = edge + 2 * NEa;         // [NV]

  k_nodes<<<(N + 255) / 256, 256, 0, stream>>>(X, batch_id, chain, V, Rall, tall, ball, call, N);
  k_global_nodes<<<1, 32, 0, stream>>>(V, Rall, tall, ball, call, N);
  k_edges<<<(E + 255) / 256, 256, 0, stream>>>(X, eidx, Rall, tall, Ef, Rts, tts, edge, E, NEa);

  // Zero the E tail: bulk via TDM (64KB tiles), remainder via async-from-LDS stores.
  float* Etail = Ef + (long long)E * 196;
  long long total_f = NEg * 196;
  int numTiles = (int)(total_f / TDM_TILE_ELTS);
  long long rem_f = total_f - (long long)numTiles * TDM_TILE_ELTS;  // multiple of 4
  if (numTiles > 0) {
    int tdmBlocks = numTiles < 64 ? numTiles : 64;
    k_zero_tdm<<<tdmBlocks, 256, 0, stream>>>(Etail, numTiles);
  }
  if (rem_f > 0) {
    k_zero_async<<<16, 256, 0, stream>>>(Etail + (long long)numTiles * TDM_TILE_ELTS, rem_f / 4);
  }

  k_global_edges<<<(int)((NEg + 255) / 256), 256, 0, stream>>>(batch_id, Rts, tts, edge, N, E, NEa);
}